// OOTGSetConvEncoder_57397942944248
// MI455X (gfx1250) — compile-verified
//
#include <hip/hip_runtime.h>
#include <hip/hip_bf16.h>

// Shapes from the reference
#define BATCH   8
#define NGRID   4096
#define NOFF    4096
#define DIM     2
#define DY      2

#define TK      256           // off-grid points per LDS chunk (double-buffered)
#define NCHUNK  (NOFF / TK)   // 16
#define ROWS_PER_BLOCK 128    // 8 waves x 16-row M tiles
#define BLOCK_THREADS  256

typedef float v2f __attribute__((ext_vector_type(2)));
typedef float v8f __attribute__((ext_vector_type(8)));

// w = exp(-0.5*sum((d/ls)^2)) = exp2(-((d*a0)^2 + (d*a1)^2)),  a_j = sqrt(0.5*log2e)/ls_j
#define SQRT_HALF_LOG2E 0.84932180028801904272150283410289f

__global__ __launch_bounds__(BLOCK_THREADS)
void ootg_setconv_wmma_kernel(const float* __restrict__ xc_off,   // [B,NOFF,2]
                              const float* __restrict__ yc_off,   // [B,NOFF,2]
                              const float* __restrict__ xc_on,    // [B,NGRID,2]
                              const float* __restrict__ yc_on,    // [B,NGRID,2]
                              const float* __restrict__ ls_param, // [2]
                              float* __restrict__ out_xc,         // [B,NGRID,2]
                              float* __restrict__ out_yc)         // [B,NGRID,4]
{
    // double-buffered staged chunks; x is pre-scaled by (a0,a1)
    __shared__ float4 sxbuf[2][TK / 2];   // 2 points per float4
    __shared__ float4 sybuf[2][TK / 2];

    const int tid   = threadIdx.x;
    const int lane  = tid & 31;
    const int wave  = tid >> 5;
    const int blk   = blockIdx.x;
    const int tilesPerBatch = NGRID / ROWS_PER_BLOCK;   // 32
    const int b     = blk / tilesPerBatch;
    const int r0    = (blk % tilesPerBatch) * ROWS_PER_BLOCK;

    // lengthscale: ls = 1e-5 + softplus(p); a_j = sqrt(0.5*log2e)/ls_j
    float p0 = ls_param[0], p1 = ls_param[1];
    float sp0 = fmaxf(p0, 0.0f) + log1pf(expf(-fabsf(p0)));
    float sp1 = fmaxf(p1, 0.0f) + log1pf(expf(-fabsf(p1)));
    float a0 = SQRT_HALF_LOG2E / (1e-5f + sp0);
    float a1 = SQRT_HALF_LOG2E / (1e-5f + sp1);

    // ---- coalesced copies: out_xc = xc_on ; out_yc[:, :, 0:2] = yc_on ----
    {
        const int base2 = (b * NGRID + r0) * 2;         // 128 rows * 2 floats
        out_xc[base2 + tid] = xc_on[base2 + tid];
        const int r = tid >> 1, c = tid & 1;
        out_yc[(b * NGRID + r0 + r) * 4 + c] = yc_on[(b * NGRID + r0 + r) * 2 + c];
    }

    // ---- this wave's 16-row M tile (pre-scaled grid point) ----
    const int m    = lane & 15;
    const int mrow = r0 + wave * 16 + m;
    const float2* xon2 = (const float2*)xc_on;
    const float2 xg_raw = xon2[b * NGRID + mrow];
    const float ugx = xg_raw.x * a0;
    const float ugy = xg_raw.y * a1;

    const float4* xoff4 = (const float4*)(xc_off) + b * (NOFF / 2);
    const float4* yoff4 = (const float4*)(yc_off) + b * (NOFF / 2);

    const int n   = lane & 15;                 // B/C column index
    const int khi = lane >> 4;                 // 0 or 1: selects K pair within step

    v8f acc = {};                              // C/D accumulator (16x16 f32)

    // ---- prologue: stage chunk 0 ----
    float4 gx, gy;
    if (tid < TK / 2) gx = xoff4[tid];
    else              gy = yoff4[tid - TK / 2];
    if (tid < TK / 2) {
        float4 s = { gx.x * a0, gx.y * a1, gx.z * a0, gx.w * a1 };
        sxbuf[0][tid] = s;
    } else {
        sybuf[0][tid - TK / 2] = gy;
    }
    __syncthreads();

    for (int c = 0; c < NCHUNK; ++c) {
        const int cur = c & 1, nxt = cur ^ 1;
        const bool hasNext = (c + 1) < NCHUNK;

        // issue next chunk's global loads early (latency hidden by compute)
        if (hasNext) {
            const int kb4 = (c + 1) * (TK / 2);
            if (tid < TK / 2) gx = xoff4[kb4 + tid];
            else              gy = yoff4[kb4 + (tid - TK / 2)];
            if (c + 2 < NCHUNK) {
                __builtin_prefetch((const char*)&xoff4[(c + 2) * (TK / 2)], 0, 0);
                __builtin_prefetch((const char*)&yoff4[(c + 2) * (TK / 2)], 0, 0);
            }
        }

        // ---- compute on buffer[cur]: 64 WMMAs per chunk ----
        const float4* __restrict__ sx4 = sxbuf[cur];
        const float4* __restrict__ sy4 = sybuf[cur];
        #pragma unroll 8
        for (int kk = 0; kk < TK; kk += 4) {
            const int q = (kk >> 1) + khi;     // float4 index: points k0, k0+1
            // A fragment: 16x4 f32 weights, 2 VGPRs/lane (one ds_load_b128)
            float4 xq = sx4[q];
            float d00 = ugx - xq.x, d01 = ugy - xq.y;   // point k0
            float d10 = ugx - xq.z, d11 = ugy - xq.w;   // point k0+1
            v2f a;
            a.x = __builtin_amdgcn_exp2f(-(d00 * d00 + d01 * d01));
            a.y = __builtin_amdgcn_exp2f(-(d10 * d10 + d11 * d11));

            // B fragment: 4x16 f32, rows = yc_off, cols 0..1 valid, rest 0
            float4 yq = sy4[q];
            v2f bb;
            bb.x = (n == 0) ? yq.x : ((n == 1) ? yq.y : 0.0f);
            bb.y = (n == 0) ? yq.z : ((n == 1) ? yq.w : 0.0f);

            // D = A x B + C   (v_wmma_f32_16x16x4_f32)
            acc = __builtin_amdgcn_wmma_f32_16x16x4_f32(
                false, a, false, bb, (short)0, acc, false, false);
        }

        // ---- stage next chunk into the other buffer ----
        if (hasNext) {
            if (tid < TK / 2) {
                float4 s = { gx.x * a0, gx.y * a1, gx.z * a0, gx.w * a1 };
                sxbuf[nxt][tid] = s;
            } else {
                sybuf[nxt][tid - TK / 2] = gy;
            }
        }
        __syncthreads();
    }

    // ---- write the two useful C columns ----
    // C layout: VGPR r, lanes 0-15 -> (M=r, N=lane); lanes 16-31 -> (M=r+8, N=lane-16)
    const int rowbase = b * NGRID + r0 + wave * 16;
    if (lane < 2) {
        const int col = 2 + lane;
        #pragma unroll
        for (int r = 0; r < 8; ++r)
            out_yc[(rowbase + r) * 4 + col] = acc[r];
    } else if (lane >= 16 && lane < 18) {
        const int col = 2 + (lane - 16);
        #pragma unroll
        for (int r = 0; r < 8; ++r)
            out_yc[(rowbase + 8 + r) * 4 + col] = acc[r];
    }
}

extern "C" void kernel_launch(void* const* d_in, const int* in_sizes, int n_in,
                              void* d_out, int out_size, void* d_ws, size_t ws_size,
                              hipStream_t stream) {
    const float* xc_off = (const float*)d_in[0];
    const float* yc_off = (const float*)d_in[1];
    const float* xc_on  = (const float*)d_in[2];
    const float* yc_on  = (const float*)d_in[3];
    const float* ls_p   = (const float*)d_in[4];

    float* out_xc = (float*)d_out;                          // [B,NGRID,2]
    float* out_yc = (float*)d_out + BATCH * NGRID * DIM;    // [B,NGRID,4]

    const int grid = BATCH * (NGRID / ROWS_PER_BLOCK);      // 256 blocks
    ootg_setconv_wmma_kernel<<<grid, BLOCK_THREADS, 0, stream>>>(
        xc_off, yc_off, xc_on, yc_on, ls_p, out_xc, out_yc);
}